// WeightedGATLayer_26620207301237
// MI455X (gfx1250) — compile-verified
//
#include <hip/hip_runtime.h>
#include <hip/hip_bf16.h>

typedef __attribute__((ext_vector_type(2))) float v2f;
typedef __attribute__((ext_vector_type(8))) float v8f;

#define IN_C   512
#define HC     256      // HEADS*OUT_C
#define HEADS  4
#define NEG_SLOPE 0.2f
#define SX_PITCH 516    // 512 + 4 pad -> pitch % 64 == 4, conflict-free b64 LDS reads

// ---------- order-preserving float <-> uint for atomicMax-based segment max ----------
__device__ __forceinline__ unsigned f2ord(float f) {
    unsigned u = __float_as_uint(f);
    return (u & 0x80000000u) ? ~u : (u | 0x80000000u);
}
__device__ __forceinline__ float ord2f(unsigned u) {
    return (u & 0x80000000u) ? __uint_as_float(u ^ 0x80000000u) : __uint_as_float(~u);
}
#define ORD_NEG_INF 0x007FFFFFu   // f2ord(-inf)

// ---------- 0: init out=bias, s=0, m=enc(-inf) ----------
__global__ void gat_init_kernel(float* __restrict__ out, const float* __restrict__ bias,
                                float* __restrict__ ssum, unsigned* __restrict__ menc,
                                int ntot, int nh) {
    int i = blockIdx.x * blockDim.x + threadIdx.x;
    if (i < ntot) out[i] = bias[i & (HC - 1)];
    if (i < nh) { ssum[i] = 0.0f; menc[i] = ORD_NEG_INF; }
}

// ---------- 1: h = x @ W^T via V_WMMA_F32_16X16X4_F32 ----------
// block = 256 threads = 8 waves; block computes rows [row0,row0+16) x all 256 cols.
// wave w owns cols [32w, 32w+32) as two 16x16 WMMA tiles, K-loop over 512 in steps of 4.
__global__ __launch_bounds__(256) void gat_gemm_kernel(const float* __restrict__ x,
                                                       const float* __restrict__ W,
                                                       float* __restrict__ hbuf,
                                                       int N) {
    __shared__ float sx[16 * SX_PITCH];
    const int row0 = blockIdx.x * 16;
    const int tid  = threadIdx.x;

    // stage 16x512 x-strip into LDS (coalesced; 32 loads/thread)
    for (int r = 0; r < 16; ++r) {
        int gr = row0 + r;
        for (int c = tid; c < IN_C; c += 256)
            sx[r * SX_PITCH + c] = (gr < N) ? x[gr * IN_C + c] : 0.0f;
    }
    __syncthreads();

    const int w    = tid >> 5;
    const int lane = tid & 31;
    const int half = lane >> 4;     // 0: lanes 0-15 (K=k,k+1), 1: lanes 16-31 (K=k+2,k+3)
    const int l    = lane & 15;
    const int c0   = w * 32;

    const float* Wr0 = W + (c0 + l) * IN_C;        // B row for tile 0 (N = c0+l)
    const float* Wr1 = W + (c0 + 16 + l) * IN_C;   // B row for tile 1

    v8f acc0 = {};
    v8f acc1 = {};

#pragma unroll 8
    for (int k = 0; k < IN_C; k += 4) {
        const int ka = k + half * 2;
        v2f a, b0, b1;
        a.x  = sx[l * SX_PITCH + ka];
        a.y  = sx[l * SX_PITCH + ka + 1];
        b0.x = Wr0[ka];
        b0.y = Wr0[ka + 1];
        b1.x = Wr1[ka];
        b1.y = Wr1[ka + 1];
        acc0 = __builtin_amdgcn_wmma_f32_16x16x4_f32(false, a, false, b0, (short)0, acc0, false, false);
        acc1 = __builtin_amdgcn_wmma_f32_16x16x4_f32(false, a, false, b1, (short)0, acc1, false, false);
    }

    // C/D layout: VGPR r -> M=r (lanes 0-15) / M=r+8 (lanes 16-31), N = lane&15
#pragma unroll
    for (int r = 0; r < 8; ++r) {
        int m  = r + half * 8;
        int gr = row0 + m;
        if (gr < N) {
            hbuf[gr * HC + c0 + l]      = acc0[r];
            hbuf[gr * HC + c0 + 16 + l] = acc1[r];
        }
    }
}

// ---------- 2: per-node attention logits a_src/a_dst [N,4] ----------
__global__ __launch_bounds__(256) void gat_attn_kernel(const float* __restrict__ hbuf,
                                                       const float* __restrict__ att_src,
                                                       const float* __restrict__ att_dst,
                                                       float* __restrict__ a_src,
                                                       float* __restrict__ a_dst) {
    __shared__ float rs[256];
    __shared__ float rd[256];
    const int n   = blockIdx.x;
    const int tid = threadIdx.x;
    float hv = hbuf[n * HC + tid];
    rs[tid] = hv * att_src[tid];   // att_* flat [H*C] matches h row layout
    rd[tid] = hv * att_dst[tid];
    __syncthreads();
    for (int off = 32; off >= 1; off >>= 1) {
        if ((tid & 63) < off) { rs[tid] += rs[tid + off]; rd[tid] += rd[tid + off]; }
        __syncthreads();
    }
    if ((tid & 63) == 0) {
        int head = tid >> 6;
        a_src[n * HEADS + head] = rs[tid];
        a_dst[n * HEADS + head] = rd[tid];
    }
}

// ---------- 3: edge logits (leaky_relu * ew), segment max via atomicMax on encoded uint ----------
__global__ void gat_edge_logit_kernel(const int* __restrict__ ei, const float* __restrict__ ew,
                                      const float* __restrict__ a_src, const float* __restrict__ a_dst,
                                      float* __restrict__ alpha, unsigned* __restrict__ menc,
                                      int E, int E4) {
    int i = blockIdx.x * blockDim.x + threadIdx.x;   // e*4 + head
    if (i >= E4) return;
    int e = i >> 2, hd = i & 3;
    int s = ei[e], d = ei[E + e];
    float v = a_src[s * HEADS + hd] + a_dst[d * HEADS + hd];
    v = (v > 0.0f) ? v : NEG_SLOPE * v;
    v *= ew[e];
    alpha[i] = v;
    atomicMax(&menc[d * HEADS + hd], f2ord(v));
}

// ---------- 4: ex = exp(alpha - m[dst]); segment sum ----------
__global__ void gat_edge_exp_kernel(const int* __restrict__ ei,
                                    float* __restrict__ alpha,
                                    const unsigned* __restrict__ menc,
                                    float* __restrict__ ssum,
                                    int E, int E4) {
    int i = blockIdx.x * blockDim.x + threadIdx.x;
    if (i >= E4) return;
    int e = i >> 2, hd = i & 3;
    int d = ei[E + e];
    float m  = ord2f(menc[d * HEADS + hd]);   // finite for every node that is a dst
    float ex = expf(alpha[i] - m);
    alpha[i] = ex;
    atomicAdd(&ssum[d * HEADS + hd], ex);
}

// ---------- 5: weighted scatter-add aggregation (block per edge, thread per channel) ----------
__global__ __launch_bounds__(256) void gat_aggregate_kernel(const int* __restrict__ ei,
                                                            const float* __restrict__ alpha,
                                                            const float* __restrict__ ssum,
                                                            const float* __restrict__ hbuf,
                                                            float* __restrict__ out,
                                                            int E) {
    const int e  = blockIdx.x;
    const int c  = threadIdx.x;
    const int hd = c >> 6;
    const int s  = ei[e];
    const int d  = ei[E + e];
    float wgt = alpha[e * HEADS + hd] / (ssum[d * HEADS + hd] + 1e-16f);
    atomicAdd(&out[d * HC + c], hbuf[s * HC + c] * wgt);
}

extern "C" void kernel_launch(void* const* d_in, const int* in_sizes, int n_in,
                              void* d_out, int out_size, void* d_ws, size_t ws_size,
                              hipStream_t stream) {
    const float* x       = (const float*)d_in[0];
    const int*   ei      = (const int*)d_in[1];
    const float* ew      = (const float*)d_in[2];
    const float* W       = (const float*)d_in[3];
    const float* att_src = (const float*)d_in[4];
    const float* att_dst = (const float*)d_in[5];
    const float* bias    = (const float*)d_in[6];
    float* out = (float*)d_out;

    const int N = in_sizes[0] / IN_C;
    const int E = in_sizes[1] / 2;

    // workspace carve-up (all sizes 256B-aligned for these N/E)
    char* ws = (char*)d_ws;
    float*    hbuf  = (float*)ws;    ws += (size_t)N * HC * sizeof(float);     // 51.2 MB
    float*    a_src = (float*)ws;    ws += (size_t)N * HEADS * sizeof(float);
    float*    a_dst = (float*)ws;    ws += (size_t)N * HEADS * sizeof(float);
    float*    ssum  = (float*)ws;    ws += (size_t)N * HEADS * sizeof(float);
    unsigned* menc  = (unsigned*)ws; ws += (size_t)N * HEADS * sizeof(unsigned);
    float*    alpha = (float*)ws;                                              // E*4 floats

    const int ntot = N * HC;
    const int nh   = N * HEADS;
    const int E4   = E * HEADS;

    gat_init_kernel<<<(ntot + 255) / 256, 256, 0, stream>>>(out, bias, ssum, menc, ntot, nh);
    gat_gemm_kernel<<<(N + 15) / 16, 256, 0, stream>>>(x, W, hbuf, N);
    gat_attn_kernel<<<N, 256, 0, stream>>>(hbuf, att_src, att_dst, a_src, a_dst);
    gat_edge_logit_kernel<<<(E4 + 255) / 256, 256, 0, stream>>>(ei, ew, a_src, a_dst, alpha, menc, E, E4);
    gat_edge_exp_kernel<<<(E4 + 255) / 256, 256, 0, stream>>>(ei, alpha, menc, ssum, E, E4);
    gat_aggregate_kernel<<<E, 256, 0, stream>>>(ei, alpha, ssum, hbuf, out, E);
}